// WaveletDenoiser_88708254531718
// MI455X (gfx1250) — compile-verified
//
#include <hip/hip_runtime.h>
#include <hip/hip_bf16.h>

typedef float v2f __attribute__((ext_vector_type(2)));
typedef float v8f __attribute__((ext_vector_type(8)));

// db8 filter h; DEC_LO = h[::-1], DEC_HI = -sgn*h, REC_LO = h, REC_HI = sgn*h[::-1]
constexpr float kH[16] = {
    0.05441584224308161f,  0.3128715909144659f,   0.6756307362980128f,
    0.5853546836548691f,  -0.015829105256023893f, -0.2840155429624281f,
    0.00047248457399797254f, 0.128747426620186f,  -0.01736930100202211f,
   -0.04408825393106472f,  0.013981027917015516f,  0.008746094047015655f,
   -0.00487035299301066f, -0.0003917403729959771f, 0.0006754494059985568f,
   -0.00011747678400228192f };

// Correlation taps (pre-reversed, as used by jax.lax.conv):
__device__ __forceinline__ constexpr float decLo(int t) { return kH[t]; }                          // DEC_LO[::-1]
__device__ __forceinline__ constexpr float decHi(int t) { return (t & 1) ? -kH[15 - t] : kH[15 - t]; } // DEC_HI[::-1]
__device__ __forceinline__ constexpr float recLo(int t) { return kH[15 - t]; }                     // REC_LO[::-1]
__device__ __forceinline__ constexpr float recHi(int t) { return (t & 1) ? kH[t] : -kH[t]; }       // REC_HI[::-1]

// ext1[j] == src[reflect(j-14)], symmetric pad of 15 with first element dropped.
// Branchless: negative p -> ~p (= -p-1), then fold over right edge with min.
__device__ __forceinline__ float reflectRead(const float* __restrict__ s, int L, int p) {
    int q = p ^ (p >> 31);          // p<0 ? -p-1 : p
    q = min(q, 2 * L - 1 - q);      // p>=L ? 2L-1-p : p
    return s[q];
}

// One DWT level via polyphase Toeplitz WMMA:
//   D[M][(f,s)] = out_f[base + 8*M + s],  M=0..15, s=0..7, f in {lo,hi}
//   A[M][tau]   = ext1[2*base + 16*M + tau],  tau = 0..31 (8 chunks of K=4)
//   B[tau][(f,s)] = K_f[tau - 2*s]  (0 outside [0,16))
// VGPR layouts (ISA 7.12.2): A: lanes0-15 rows, VGPR0/1 = K{0,1}; lanes16-31 K{2,3}.
// B: VGPR0 rows {0,2}, VGPR1 rows {1,3}; N = lane&15.  D: VGPR v -> M = v + (lane>=16?8:0).
// Destinations are padded to a 128-float multiple, so all 256 lanes*vals store unconditionally.
__device__ __forceinline__ void dwtLevel(const float* __restrict__ src, int L,
                                         float* __restrict__ aDst, float* __restrict__ dDst,
                                         int nOut, int wave, int lane, int nWaves) {
    const int  nGroups = (nOut + 127) >> 7;
    const int  Mr      = lane & 15;           // A row index
    const bool hiHalf  = lane >= 16;
    const int  rofs    = hiHalf ? 2 : 0;      // K offset within a chunk
    const int  n       = lane & 15;           // B/D column
    const int  s       = n & 7;               // filter shift
    const int  f       = n >> 3;              // 0 = lo, 1 = hi

    // Hoist B (lane-constant): bx[c] = B[4c+rofs][(f,s)], by[c] = B[4c+rofs+1][(f,s)]
    float bx[8], by[8];
#pragma unroll
    for (int c = 0; c < 8; ++c) {
        const int k0 = 4 * c + rofs - 2 * s;
        bx[c] = ((unsigned)k0 < 16u)       ? (f ? decHi(k0)     : decLo(k0))     : 0.f;
        by[c] = ((unsigned)(k0 + 1) < 16u) ? (f ? decHi(k0 + 1) : decLo(k0 + 1)) : 0.f;
    }
    float* const myDst = (f ? dDst : aDst) + (hiHalf ? 64 : 0) + s;  // lane-constant base

    for (int g = wave; g < nGroups; g += nWaves) {
        const int base  = g << 7;             // 128 outputs per filter per group
        const int xbase = 2 * base + 16 * Mr; // tau=0 position in ext1 for this row
        v2f a[8];
        // Interior iff every p = xbase' + tau - 14 over all lanes/taus lands in [0, L):
        // min p = 2*base - 14, max p = 2*base + 240 + 31 - 14 = 2*base + 257.
        // Condition is wave-uniform; readfirstlane so the compiler emits a scalar branch.
        const int interior = __builtin_amdgcn_readfirstlane(
            ((2 * base >= 14) && (2 * base + 257 <= L - 1)) ? 1 : 0);
        if (interior) {
            const float* p0 = src + (xbase + rofs - 14);   // even float index -> 8B aligned
#pragma unroll
            for (int c = 0; c < 8; ++c)
                a[c] = *reinterpret_cast<const v2f*>(p0 + 4 * c);   // ds_load_b64 pairs
        } else {
#pragma unroll
            for (int c = 0; c < 8; ++c) {
                const int p = xbase + 4 * c + rofs - 14;
                a[c].x = reflectRead(src, L, p);
                a[c].y = reflectRead(src, L, p + 1);
            }
        }
        v8f acc = {0.f, 0.f, 0.f, 0.f, 0.f, 0.f, 0.f, 0.f};
#pragma unroll
        for (int c = 0; c < 8; ++c) {
            v2f br; br.x = bx[c]; br.y = by[c];
            acc = __builtin_amdgcn_wmma_f32_16x16x4_f32(
                      false, a[c], false, br, (short)0, acc, false, false);
        }
#pragma unroll
        for (int v = 0; v < 8; ++v)
            myDst[base + 8 * v] = acc[v];      // unconditional: dest padded past nOut
    }
}

// One IDWT level: rec[j] = sum over taps t with (j+t) odd of recLo(t)*ca[q] + recHi(t)*cd[q],
// q = (j+t-1)/2.  Outputs j=2p and j=2p+1 share the same 8 consecutive q = p..p+7
// (max q = nIn-1, so no bounds checks needed).  One aligned v2f store per pair.
__device__ __forceinline__ void idwtLevel(const float* __restrict__ ca,
                                          const float* __restrict__ cd, int nIn,
                                          float* __restrict__ dst, int tid, int nThreads) {
    const int outLen = 2 * nIn - 14;          // always even
    for (int p = tid; (p << 1) < outLen; p += nThreads) {
        float accE = 0.f, accO = 0.f;
#pragma unroll
        for (int u = 0; u < 8; ++u) {
            const float cav = ca[p + u];
            const float cdv = cd[p + u];
            accE = fmaf(recLo(2 * u + 1), cav, fmaf(recHi(2 * u + 1), cdv, accE)); // even j
            accO = fmaf(recLo(2 * u),     cav, fmaf(recHi(2 * u),     cdv, accO)); // odd j
        }
        v2f o; o.x = accE; o.y = accO;
        *reinterpret_cast<v2f*>(dst + (p << 1)) = o;   // ds_store_b64
    }
}

__device__ __forceinline__ void softThr(float* __restrict__ d, int n, float thr, int tid) {
    for (int i = tid; i < n; i += 256) {
        const float v = d[i];
        const float a = fabsf(v) - thr;
        d[i] = (a > 0.f) ? copysignf(a, v) : 0.f;
    }
}

// Level lengths for L=16384: n_{k+1} = (n_k + 15) >> 1
#define L0 16384
#define N1 8199
#define N2 4107
#define N3 2061
#define N4 1038
// LDS layout (floats); every DWT destination padded to a 128 multiple (full group)
#define OFF_B    16384        // size 8320  (a1/a3/r4/r2; r2 needs 8200)
#define OFF_D1   24704        // size 8320
#define OFF_D2   33024        // size 4224
#define OFF_D3   37248        // size 2176
#define OFF_D4   39424        // size 1152
#define OFF_HIST 40576        // 256 ints, 16B aligned
#define SMEM_BYTES 163840     // 160KB -> exactly 2 WGs per 320KB WGP

extern "C" __global__ __launch_bounds__(256)
void wavelet_denoise_kernel(const float* __restrict__ x, float* __restrict__ out) {
    extern __shared__ float smem[];
    float* A  = smem;              // 16384: a0 / a2 / a4 / r3(@+8192) / r1(final)
    float* Bb = smem + OFF_B;      // a1 / a3 / r4 / r2
    float* d1 = smem + OFF_D1;
    float* d2 = smem + OFF_D2;
    float* d3 = smem + OFF_D3;
    float* d4 = smem + OFF_D4;
    int*   hist = reinterpret_cast<int*>(smem + OFF_HIST); // 256
    int*   ctrl = hist + 256;                              // [0]=prefix bits, [1]=k remaining

    const int    tid    = threadIdx.x;
    const int    lane   = tid & 31;
    const int    wave   = tid >> 5;
    const size_t rowOff = (size_t)blockIdx.x * L0;

    // ---- load row into LDS (float4 vectorized) ----
    const float4* xr = reinterpret_cast<const float4*>(x + rowOff);
    float4*       A4 = reinterpret_cast<float4*>(A);
    for (int i = tid; i < L0 / 4; i += 256) A4[i] = xr[i];
    __syncthreads();

    // ---- forward DWT, 4 levels (WMMA f32 16x16x4, polyphase tiling) ----
    dwtLevel(A,  L0, Bb, d1, N1, wave, lane, 8); __syncthreads();
    dwtLevel(Bb, N1, A,  d2, N2, wave, lane, 8); __syncthreads();
    dwtLevel(A,  N2, Bb, d3, N3, wave, lane, 8); __syncthreads();
    dwtLevel(Bb, N3, A,  d4, N4, wave, lane, 8); __syncthreads();

    // ---- exact median of |d1| via 4-pass radix select on float bits ----
    if (tid == 0) { ctrl[0] = 0; ctrl[1] = (N1 - 1) / 2; }  // k = 4099 (odd count -> middle)
    __syncthreads();
    for (int byte = 3; byte >= 0; --byte) {
        hist[tid] = 0;
        __syncthreads();
        const unsigned prefix = (unsigned)ctrl[0];
        for (int i = tid; i < N1; i += 256) {
            const unsigned u = __float_as_uint(fabsf(d1[i]));  // >=0: bit order == value order
            const bool match = (byte == 3) ? true : ((u >> ((byte + 1) * 8)) == prefix);
            if (match) atomicAdd(&hist[(u >> (byte * 8)) & 255], 1);
        }
        __syncthreads();
        if (tid == 0) {
            int k = ctrl[1], cum = 0, b = 0;
            for (; b < 255; ++b) { const int h = hist[b]; if (cum + h > k) break; cum += h; }
            ctrl[1] = k - cum;
            ctrl[0] = (int)((prefix << 8) | (unsigned)b);
        }
        __syncthreads();
    }
    const float median = __uint_as_float((unsigned)ctrl[0]);
    const float thr    = (median / 0.6745f) * 4.4054651f;   // sqrt(2*ln(16384))

    // ---- soft-threshold all detail bands in place ----
    softThr(d1, N1, thr, tid);
    softThr(d2, N2, thr, tid);
    softThr(d3, N3, thr, tid);
    softThr(d4, N4, thr, tid);
    __syncthreads();

    // ---- inverse DWT, 4 levels (with truncation to detail length) ----
    idwtLevel(A,        d4, N4, Bb,       tid, 256); __syncthreads(); // 2062 -> Bb
    idwtLevel(Bb,       d3, N3, A + 8192, tid, 256); __syncthreads(); // 4108 -> A[8192..]
    idwtLevel(A + 8192, d2, N2, Bb,       tid, 256); __syncthreads(); // 8200 -> Bb
    idwtLevel(Bb,       d1, N1, A,        tid, 256); __syncthreads(); // 16384 -> A

    // ---- store row ----
    float4* o4 = reinterpret_cast<float4*>(out + rowOff);
    for (int i = tid; i < L0 / 4; i += 256) o4[i] = A4[i];
}

extern "C" void kernel_launch(void* const* d_in, const int* in_sizes, int n_in,
                              void* d_out, int out_size, void* d_ws, size_t ws_size,
                              hipStream_t stream) {
    (void)n_in; (void)out_size; (void)d_ws; (void)ws_size;
    const float* xp  = (const float*)d_in[0];
    float*       op  = (float*)d_out;
    const int    rows = in_sizes[0] / L0;   // 128*16 = 2048

    // Raise dynamic-LDS cap (CDNA5 WGP has 320KB; we use 160KB/WG -> 2 WGs/WGP).
    (void)hipFuncSetAttribute(reinterpret_cast<const void*>(wavelet_denoise_kernel),
                              hipFuncAttributeMaxDynamicSharedMemorySize, SMEM_BYTES);

    wavelet_denoise_kernel<<<dim3(rows), dim3(256), SMEM_BYTES, stream>>>(xp, op);
}